// KeypointDetction_9706626089753
// MI455X (gfx1250) — compile-verified
//
#include <hip/hip_runtime.h>
#include <stdint.h>

typedef __attribute__((ext_vector_type(2))) float v2f;
typedef __attribute__((ext_vector_type(8))) float v8f;

#define Q_LEVEL 0.3f
#define CAP 8192
#define SORT_N 8192

__device__ __forceinline__ float band7(int m, int k) {
  // 1.0 iff m <= k <= m+6  (7-tap box band)
  return (k >= m && k <= m + 6) ? 1.0f : 0.0f;
}

// ---- CDNA5 async global->LDS copies (ASYNCcnt-tracked), inline asm form ----
__device__ __forceinline__ void async_g2l_b32(uint32_t ldsOff, const void* gaddr) {
  asm volatile("global_load_async_to_lds_b32 %0, %1, off"
               :: "v"(ldsOff), "v"(gaddr) : "memory");
}
__device__ __forceinline__ void async_g2l_b64(uint32_t ldsOff, const void* gaddr) {
  asm volatile("global_load_async_to_lds_b64 %0, %1, off"
               :: "v"(ldsOff), "v"(gaddr) : "memory");
}
__device__ __forceinline__ void wait_async0() {
  asm volatile("s_wait_asynccnt 0x0" ::: "memory");
}

// ---------------------------------------------------------------- init
__global__ void k_zero_out(float4* __restrict__ out, long n4) {
  long i = (long)blockIdx.x * blockDim.x + threadIdx.x;
  if (i < n4) out[i] = make_float4(0.f, 0.f, 0.f, 0.f);
}

__global__ void k_init_small(int* __restrict__ cnt, int* __restrict__ rmax, int B) {
  int t = threadIdx.x;
  if (t < B) { cnt[t] = 0; rmax[t] = 0; }
}

// ------------------------------------------------- response (WMMA box filter)
// One 128-thread block (4 waves) computes a 32x32 tile of the Shi-Tomasi
// response. Sobel + products in VALU; the three 7x7 box sums run on the
// f32 matrix pipe as banded-matrix WMMAs (V * P * H).
__global__ __launch_bounds__(128) void k_response(const float* __restrict__ im,
                                                  float* __restrict__ R,
                                                  int* __restrict__ rmax,
                                                  int H, int W) {
  __shared__ float simg[40][41];      // 32x32 tile + halo 4 (Sobel+box)
  __shared__ float sP[3][38][40];     // Ixx/Iyy/Ixy, 38x38 valid + 2 zero pad cols
  __shared__ float sQ[4][24][16];     // per-wave stage-1 output (22 rows + 2 zero)
  __shared__ int smax;

  const int tid = threadIdx.x;
  const int b  = blockIdx.z;
  const int r0 = blockIdx.y * 32;
  const int c0 = blockIdx.x * 32;
  const float* img = im + (size_t)b * H * W;

  if (tid == 0) smax = 0;

  // Load 40x40 input tile (halo 4), emulate uint8 quantization; zero pad OOB.
  for (int i = tid; i < 40 * 40; i += 128) {
    int rr = i / 40, cc = i % 40;
    int gr = r0 - 4 + rr, gc = c0 - 4 + cc;
    float v = 0.f;
    if (gr >= 0 && gr < H && gc >= 0 && gc < W) {
      float x = img[(size_t)gr * W + gc];
      v = fminf(fmaxf(floorf(x * 255.f), 0.f), 255.f) * (1.f / 255.f);
    }
    simg[rr][cc] = v;
  }
  __syncthreads();

  // Sobel gradients + structure-tensor products over the 38x38 region
  // (output 32x32 + halo 3). Cols 38..39 are explicit zeros (WMMA K padding:
  // must be 0, not garbage, to avoid 0*NaN poisoning).
  for (int i = tid; i < 38 * 40; i += 128) {
    int p = i / 40, q = i % 40;
    float xx = 0.f, yy = 0.f, xy = 0.f;
    int gr = r0 - 3 + p, gc = c0 - 3 + q;
    if (q < 38 && gr >= 0 && gr < H && gc >= 0 && gc < W) {
      int rr = p + 1, cc = q + 1;
      float a00 = simg[rr-1][cc-1], a01 = simg[rr-1][cc], a02 = simg[rr-1][cc+1];
      float a10 = simg[rr  ][cc-1],                       a12 = simg[rr  ][cc+1];
      float a20 = simg[rr+1][cc-1], a21 = simg[rr+1][cc], a22 = simg[rr+1][cc+1];
      float ix = (a02 + 2.f * a12 + a22) - (a00 + 2.f * a10 + a20);
      float iy = (a20 + 2.f * a21 + a22) - (a00 + 2.f * a01 + a02);
      xx = ix * ix; yy = iy * iy; xy = ix * iy;
    }
    sP[0][p][q] = xx; sP[1][p][q] = yy; sP[2][p][q] = xy;
  }
  __syncthreads();

  const int w    = tid >> 5;          // wave id (wave32)
  const int lane = tid & 31;
  const int sr = (w >> 1) * 16;       // sub-tile base in product coords
  const int sc = (w & 1) * 16;
  const int halfsel = (lane < 16) ? 0 : 2;  // K-pair select per lane half
  const int n = lane & 15;            // N (or M for A fragments)

  // Band fragments are channel-invariant and identical for stage-1 B and
  // stage-2 A (V = H^T under this lane mapping): build them exactly once.
  v2f bandf[6];
  #pragma unroll
  for (int kk = 0; kk < 6; ++kk) {
    int kb = 4 * kk + halfsel;
    bandf[kk].x = band7(n, kb);
    bandf[kk].y = band7(n, kb + 1);
  }

  v8f acc[3];
  #pragma unroll
  for (int ch = 0; ch < 3; ++ch) {
    // ---- Stage 1: horizontal 7-sums  Q(22x16) = P_rows(22x24) * H(24x16)
    v8f q0 = {};   // M rows 0..15
    v8f q1 = {};   // M rows 16..21 (22..31 garbage, contained in unused D rows)
    #pragma unroll
    for (int kk = 0; kk < 6; ++kk) {
      int kb = 4 * kk + halfsel;
      int acol = sc + kb;                  // <= 39, pad cols are zero
      v2f a0;                              // A[m][k] = P[sr+m][sc+k]
      a0.x = sP[ch][sr + n][acol];
      a0.y = sP[ch][sr + n][acol + 1];
      q0 = __builtin_amdgcn_wmma_f32_16x16x4_f32(false, a0, false, bandf[kk],
                                                 (short)0, q0, false, false);
      int arow1 = sr + 16 + n; if (arow1 > 37) arow1 = 37;  // clamp; M-garbage safe
      v2f a1;
      a1.x = sP[ch][arow1][acol];
      a1.y = sP[ch][arow1][acol + 1];
      q1 = __builtin_amdgcn_wmma_f32_16x16x4_f32(false, a1, false, bandf[kk],
                                                 (short)0, q1, false, false);
    }
    // Spill Q to LDS in row-major (D layout: VGPR r <-> rows r / r+8).
    #pragma unroll
    for (int rr = 0; rr < 8; ++rr) {
      int row = (lane < 16) ? rr : rr + 8;
      sQ[w][row][n] = q0[rr];
    }
    #pragma unroll
    for (int rr = 0; rr < 6; ++rr) {
      if (lane < 16) sQ[w][16 + rr][n] = q1[rr];
    }
    if (lane < 16) sQ[w][22][n] = 0.f; else sQ[w][23][n] = 0.f;  // K padding

    // ---- Stage 2: vertical 7-sums  S(16x16) = V(16x24) * Q(24x16)
    v8f c = {};
    #pragma unroll
    for (int kk = 0; kk < 6; ++kk) {
      int kb = 4 * kk + halfsel;
      v2f bq;                              // B[k][n] = Q[k][n]
      bq.x = sQ[w][kb][n];
      bq.y = sQ[w][kb + 1][n];
      c = __builtin_amdgcn_wmma_f32_16x16x4_f32(false, bandf[kk], false, bq,
                                                (short)0, c, false, false);
    }
    acc[ch] = c;
  }

  // Min-eigenvalue response, write R, track per-image max (R >= 0: PSD tensor).
  float tmax = 0.f;
  #pragma unroll
  for (int rr = 0; rr < 8; ++rr) {
    float sxx = acc[0][rr], syy = acc[1][rr], sxy = acc[2][rr];
    float ht = 0.5f * (sxx + syy);
    float dd = 0.5f * (sxx - syy);
    float resp = ht - sqrtf(dd * dd + sxy * sxy);
    int orow = r0 + sr + ((lane < 16) ? rr : rr + 8);
    int ocol = c0 + sc + n;
    R[((size_t)b * H + orow) * W + ocol] = resp;
    tmax = fmaxf(tmax, resp);
  }
  atomicMax(&smax, __float_as_int(fmaxf(tmax, 0.f)));
  __syncthreads();
  if (tid == 0) atomicMax(&rmax[b], smax);
}

// --------------------------------------------- fused 15x15 NMS + compaction
// R tile is staged with CDNA5 async global->LDS copies (ASYNCcnt path);
// out-of-image cells get -inf via plain DS stores to disjoint addresses.
__global__ __launch_bounds__(256) void k_nms(const float* __restrict__ R,
                                             const int* __restrict__ rmax,
                                             int* __restrict__ cnt,
                                             unsigned long long* __restrict__ cand,
                                             int H, int W) {
  __shared__ float s[46][48];
  __shared__ float hm[46][32];
  const int tid = threadIdx.x;
  const int b  = blockIdx.z;
  const int r0 = blockIdx.y * 32, c0 = blockIdx.x * 32;
  const float* Rb = R + (size_t)b * H * W;
  const float NEG = -__builtin_inff();

  for (int i = tid; i < 46 * 46; i += 256) {
    int rr = i / 46, cc = i % 46;
    int gr = r0 - 7 + rr, gc = c0 - 7 + cc;
    if (gr >= 0 && gr < H && gc >= 0 && gc < W) {
      async_g2l_b32((uint32_t)(uintptr_t)&s[rr][cc], &Rb[(size_t)gr * W + gc]);
    } else {
      s[rr][cc] = NEG;
    }
  }
  wait_async0();          // each wave drains its own ASYNCcnt
  __syncthreads();        // cross-wave visibility

  for (int i = tid; i < 46 * 32; i += 256) {
    int rr = i / 32, cc = i % 32;
    float m = s[rr][cc];
    #pragma unroll
    for (int d = 1; d < 15; ++d) m = fmaxf(m, s[rr][cc + d]);
    hm[rr][cc] = m;
  }
  __syncthreads();
  float thr = Q_LEVEL * __int_as_float(rmax[b]);
  for (int i = tid; i < 32 * 32; i += 256) {
    int rr = i / 32, cc = i % 32;
    float center = s[rr + 7][cc + 7];
    float m = hm[rr][cc];
    #pragma unroll
    for (int d = 1; d < 15; ++d) m = fmaxf(m, hm[rr + d][cc]);
    if (center >= m && center >= thr) {
      int pos = atomicAdd(&cnt[b], 1);
      if (pos < CAP) {
        unsigned idx = (unsigned)((r0 + rr) * W + (c0 + cc));
        // key: score bits (>=0, monotone) high, ~idx low (smaller idx wins ties)
        unsigned long long key =
            ((unsigned long long)__float_as_uint(center) << 32) |
            (unsigned long long)(0xFFFFFFFFu - idx);
        cand[(size_t)b * CAP + pos] = key;
      }
    }
  }
}

// ---------------------------------------------- per-image top-K (bitonic)
__global__ __launch_bounds__(1024) void k_topk(const unsigned long long* __restrict__ cand,
                                               const int* __restrict__ cnt,
                                               const int* __restrict__ topk,
                                               float* __restrict__ out,
                                               int H, int W) {
  __shared__ unsigned long long keys[SORT_N];
  const int tid = threadIdx.x;
  const int b = blockIdx.x;
  int n = cnt[b]; if (n > CAP) n = CAP;
  for (int i = tid; i < SORT_N; i += 1024) {
    if (i < n) {
      async_g2l_b64((uint32_t)(uintptr_t)&keys[i], &cand[(size_t)b * CAP + i]);
    } else {
      keys[i] = 0ull;
    }
  }
  wait_async0();
  __syncthreads();
  for (unsigned k = 2; k <= (unsigned)SORT_N; k <<= 1) {
    for (unsigned j = k >> 1; j > 0; j >>= 1) {
      for (unsigned i = tid; i < (unsigned)SORT_N; i += 1024) {
        unsigned ixj = i ^ j;
        if (ixj > i) {
          unsigned long long a = keys[i], c = keys[ixj];
          bool swap = ((i & k) == 0) ? (a < c) : (a > c);  // descending sort
          if (swap) { keys[i] = c; keys[ixj] = a; }
        }
      }
      __syncthreads();
    }
  }
  int K = topk[0];
  int lim = (n < K) ? n : K;
  for (int j = tid; j < lim; j += 1024) {
    unsigned idx = 0xFFFFFFFFu - (unsigned)(keys[j] & 0xFFFFFFFFu);
    out[(size_t)b * H * W + idx] = 1.0f;
  }
}

// ---------------------------------------------------------------- launch
extern "C" void kernel_launch(void* const* d_in, const int* in_sizes, int n_in,
                              void* d_out, int out_size, void* d_ws, size_t ws_size,
                              hipStream_t stream) {
  const float* im  = (const float*)d_in[0];
  const int* topk  = (const int*)d_in[1];
  float* out = (float*)d_out;
  const int H = 1024, W = 1024;
  const int B = in_sizes[0] / (H * W);

  char* ws = (char*)d_ws;
  size_t rBytes = (size_t)B * H * W * sizeof(float);
  float* Rmap = (float*)ws;
  int* rmax   = (int*)(ws + rBytes);
  int* cnt    = (int*)(ws + rBytes + 256);
  unsigned long long* cand = (unsigned long long*)(ws + rBytes + 512);

  long n4 = (long)out_size / 4;
  k_zero_out<<<(int)((n4 + 255) / 256), 256, 0, stream>>>((float4*)out, n4);
  k_init_small<<<1, 64, 0, stream>>>(cnt, rmax, B);

  dim3 grid(W / 32, H / 32, B);
  k_response<<<grid, 128, 0, stream>>>(im, Rmap, rmax, H, W);
  k_nms<<<grid, 256, 0, stream>>>(Rmap, rmax, cnt, cand, H, W);
  k_topk<<<B, 1024, 0, stream>>>(cand, cnt, topk, out, H, W);
}